// MILCellModelmerge_75307956568754
// MI455X (gfx1250) — compile-verified
//
#include <hip/hip_runtime.h>
#include <math.h>

// ---------------------------------------------------------------------------
// MI455X (gfx1250) implementation of MILCellModel forward.
// All linear algebra goes through a fp32 WMMA GEMM (V_WMMA_F32_16X16X4_F32):
// the model is memory-bound (~42 GFLOP vs hundreds of MB of traffic at
// 23.3 TB/s), so full fp32 precision on the matrix pipe is the right choice.
// Requires ws_size >= ~140 MB (attention score buffer dominates).
// ---------------------------------------------------------------------------

typedef __attribute__((ext_vector_type(2))) float v2f;
typedef __attribute__((ext_vector_type(8))) float v8f;

// model dims
#define NB   8
#define NN   1024
#define NE   512
#define ND   256
#define NH   2
#define NHD  128      // head dim
#define NHID 1024
#define NNQ  4
#define NOUT 1024
#define NCLS 24
#define NCH  128      // GAT per-head channels
#define NK   8
#define NODES (NB * NN)   // 8192

// GEMM tiling
#define TM 64
#define TN 64
#define TKt 32

// ---------------------------------------------------------------------------
// WMMA fp32 GEMM:  C = alpha * A @ op(B) + bias + addend, optional ReLU.
//   A: [M,K] (row major, ld=lda), B: transB ? [N,K] : [K,N]
//   batched over gridDim.z = nOuter*nInner with separate outer/inner strides
// Block = 256 threads = 8 waves; each wave owns a 16x32 slice of the 64x64
// C tile (two 16x16 accumulators). K is consumed 32 at a time via LDS tiles,
// 8 x v_wmma_f32_16x16x4_f32 steps per accumulator per tile.
// ---------------------------------------------------------------------------
__global__ __launch_bounds__(256)
void wmma_gemm_kernel(const float* __restrict__ A, long long lda, long long sAo, long long sAi,
                      const float* __restrict__ Bp, long long ldb, long long sBo, long long sBi,
                      float* __restrict__ C, long long ldc, long long sCo, long long sCi,
                      const float* __restrict__ bias,
                      const float* __restrict__ addend, long long ldadd,
                      int M, int Nn, int Kd, int nInner,
                      float alpha, int transB, int doRelu)
{
    __shared__ float As[TM][TKt + 1];
    __shared__ float Bs[TN][TKt + 1];

    const int z = blockIdx.z;
    const int outer = z / nInner, inner = z % nInner;
    A  += outer * sAo + inner * sAi;
    Bp += outer * sBo + inner * sBi;
    C  += outer * sCo + inner * sCi;

    const int row0 = blockIdx.y * TM;
    const int col0 = blockIdx.x * TN;
    const int tid  = threadIdx.x;
    const int lane = tid & 31, wid = tid >> 5;
    const int mw = (wid & 3) * 16;        // wave's M sub-tile
    const int nw = (wid >> 2) * 32;       // wave's N group (two 16-wide frags)
    const int lh = lane >> 4, lm = lane & 15;

    v8f acc0 = {};
    v8f acc1 = {};

    const int ktiles = (Kd + TKt - 1) / TKt;
    for (int kt = 0; kt < ktiles; ++kt) {
        const int k0 = kt * TKt;
        // global -> LDS (zero-fill out of range so EXEC stays uniform)
#pragma unroll
        for (int i = 0; i < (TM * TKt) / 256; ++i) {
            int linear = tid + i * 256;
            int r = linear >> 5, c = linear & 31;
            int gr = row0 + r, gc = k0 + c;
            As[r][c] = (gr < M && gc < Kd) ? A[(long long)gr * lda + gc] : 0.f;
        }
#pragma unroll
        for (int i = 0; i < (TN * TKt) / 256; ++i) {
            int linear = tid + i * 256;
            int r = linear >> 5, c = linear & 31;
            int gn = col0 + r, gc = k0 + c;
            float v = 0.f;
            if (gn < Nn && gc < Kd)
                v = transB ? Bp[(long long)gn * ldb + gc]
                           : Bp[(long long)gc * ldb + gn];
            Bs[r][c] = v;
        }
        __syncthreads();

#pragma unroll
        for (int kk = 0; kk < TKt; kk += 4) {
            // A frag (16x4 f32): lane L holds m=L&15, k=(L>>4)*2 + v
            v2f a, b0, b1;
            a[0]  = As[mw + lm][kk + lh * 2 + 0];
            a[1]  = As[mw + lm][kk + lh * 2 + 1];
            // B frag (4x16 f32): lane L holds n=L&15, k=(L>>4)*2 + v
            b0[0] = Bs[nw + lm][kk + lh * 2 + 0];
            b0[1] = Bs[nw + lm][kk + lh * 2 + 1];
            b1[0] = Bs[nw + 16 + lm][kk + lh * 2 + 0];
            b1[1] = Bs[nw + 16 + lm][kk + lh * 2 + 1];
            acc0 = __builtin_amdgcn_wmma_f32_16x16x4_f32(
                       false, a, false, b0, (short)0, acc0, false, false);
            acc1 = __builtin_amdgcn_wmma_f32_16x16x4_f32(
                       false, a, false, b1, (short)0, acc1, false, false);
        }
        __syncthreads();
    }

    // C/D layout: VGPR r -> m = r + 8*(lane>>4), n = lane&15
#pragma unroll
    for (int r = 0; r < 8; ++r) {
        int m = mw + r + 8 * lh;
        int gm = row0 + m;
        if (gm >= M) continue;
#pragma unroll
        for (int j = 0; j < 2; ++j) {
            int gn = col0 + nw + j * 16 + lm;
            if (gn >= Nn) continue;
            float v = (j == 0) ? acc0[r] : acc1[r];
            v *= alpha;
            if (bias)   v += bias[gn];
            if (addend) v += addend[(long long)gm * ldadd + gn];
            if (doRelu) v = v > 0.f ? v : 0.f;
            C[(long long)gm * ldc + gn] = v;
        }
    }
}

// ---------------------------------------------------------------------------
// helper kernels
// ---------------------------------------------------------------------------

__device__ inline float blk_reduce_sum(float v, float* red) {
    int t = threadIdx.x;
    red[t] = v; __syncthreads();
    for (int s = 128; s > 0; s >>= 1) {
        if (t < s) red[t] += red[t + s];
        __syncthreads();
    }
    float r = red[0]; __syncthreads();
    return r;
}

__device__ inline float blk_reduce_max(float v, float* red) {
    int t = threadIdx.x;
    red[t] = v; __syncthreads();
    for (int s = 128; s > 0; s >>= 1) {
        if (t < s) red[t] = fmaxf(red[t], red[t + s]);
        __syncthreads();
    }
    float r = red[0]; __syncthreads();
    return r;
}

// gather position-embedding table rows (clip + trunc index)
__global__ __launch_bounds__(128)
void pe_gather_kernel(const float* __restrict__ pos,
                      const float* __restrict__ xt, const float* __restrict__ yt,
                      float* __restrict__ xg, float* __restrict__ yg)
{
    int node = blockIdx.x;            // 0..NODES-1
    int t = threadIdx.x;              // 0..127
    float px = pos[node * 2 + 0], py = pos[node * 2 + 1];
    int xi = (int)fminf(fmaxf(px, 0.f), 1000.f);
    int yi = (int)fminf(fmaxf(py, 0.f), 1000.f);
    xg[node * 128 + t] = xt[xi * 128 + t];
    yg[node * 128 + t] = yt[yi * 128 + t];
}

// per-batch mean/std of coordinates (population std, axis 0)
__global__ __launch_bounds__(256)
void pos_stats_kernel(const float* __restrict__ pos, float* __restrict__ stats)
{
    __shared__ float red[256];
    int b = blockIdx.x;
    float sx = 0, sy = 0, sxx = 0, syy = 0;
    for (int i = threadIdx.x; i < NN; i += 256) {
        float x = pos[(b * NN + i) * 2 + 0];
        float y = pos[(b * NN + i) * 2 + 1];
        sx += x; sy += y; sxx += x * x; syy += y * y;
    }
    sx  = blk_reduce_sum(sx, red);
    sy  = blk_reduce_sum(sy, red);
    sxx = blk_reduce_sum(sxx, red);
    syy = blk_reduce_sum(syy, red);
    if (threadIdx.x == 0) {
        float mx = sx / NN, my = sy / NN;
        float vx = fmaxf(sxx / NN - mx * mx, 0.f);
        float vy = fmaxf(syy / NN - my * my, 0.f);
        stats[b * 4 + 0] = mx;
        stats[b * 4 + 1] = my;
        stats[b * 4 + 2] = sqrtf(vx);
        stats[b * 4 + 3] = sqrtf(vy);
    }
}

// 8-NN (incl. self) over normalized coords; ascending distance, stable in j
__global__ __launch_bounds__(256)
void knn_kernel(const float* __restrict__ pos, const float* __restrict__ stats,
                int* __restrict__ nbr)
{
    int b = blockIdx.y;
    int i = blockIdx.x * blockDim.x + threadIdx.x;
    if (i >= NN) return;
    float mx = stats[b * 4 + 0], my = stats[b * 4 + 1];
    float sx = stats[b * 4 + 2] + 1e-8f, sy = stats[b * 4 + 3] + 1e-8f;
    float xi = (pos[(b * NN + i) * 2 + 0] - mx) / sx;
    float yi = (pos[(b * NN + i) * 2 + 1] - my) / sy;
    float bd[NK]; int bi[NK];
#pragma unroll
    for (int k = 0; k < NK; ++k) { bd[k] = 3.0e38f; bi[k] = 0; }
    for (int j = 0; j < NN; ++j) {
        float dx = (pos[(b * NN + j) * 2 + 0] - mx) / sx - xi;
        float dy = (pos[(b * NN + j) * 2 + 1] - my) / sy - yi;
        float d = dx * dx + dy * dy;
        if (d < bd[NK - 1]) {
            int k = NK - 1;
            while (k > 0 && bd[k - 1] > d) {
                bd[k] = bd[k - 1]; bi[k] = bi[k - 1]; --k;
            }
            bd[k] = d; bi[k] = j;
        }
    }
#pragma unroll
    for (int k = 0; k < NK; ++k) nbr[(b * NN + i) * NK + k] = bi[k];
}

// per-(node,head) attention scalars: es = <hw, as>, ed = <hw, ad>
__global__ __launch_bounds__(256)
void gat_score_kernel(const float* __restrict__ hw,
                      const float* __restrict__ as_, const float* __restrict__ ad_,
                      float* __restrict__ es, float* __restrict__ ed)
{
    int idx = blockIdx.x * blockDim.x + threadIdx.x;   // over NODES*NH
    if (idx >= NODES * NH) return;
    int h = idx & 1, node = idx >> 1;
    const float* v = hw + (long long)node * ND + h * NCH;
    float s = 0.f, d = 0.f;
    for (int c = 0; c < NCH; ++c) {
        float x = v[c];
        s += x * as_[h * NCH + c];
        d += x * ad_[h * NCH + c];
    }
    es[idx] = s; ed[idx] = d;
}

// softmax over K=8 neighbors of leaky_relu(es_j + ed_i), aggregate, bias+relu
__global__ __launch_bounds__(256)
void gat_aggregate_kernel(const float* __restrict__ hw,
                          const float* __restrict__ es, const float* __restrict__ ed,
                          const int* __restrict__ nbr, const float* __restrict__ bias,
                          float* __restrict__ out)
{
    int node = blockIdx.x * 8 + (threadIdx.x >> 5);  // one wave per node
    int lane = threadIdx.x & 31;
    if (node >= NODES) return;
    int bbase = (node >> 10) << 10;                  // batch offset in nodes
    int jg[NK];
#pragma unroll
    for (int k = 0; k < NK; ++k) jg[k] = bbase + nbr[node * NK + k];

    for (int h = 0; h < NH; ++h) {
        float edv = ed[node * NH + h];
        float w[NK]; float mxe = -3.0e38f;
#pragma unroll
        for (int k = 0; k < NK; ++k) {
            float e = es[jg[k] * NH + h] + edv;
            e = e > 0.f ? e : 0.2f * e;              // leaky_relu(0.2)
            w[k] = e; mxe = fmaxf(mxe, e);
        }
        float ssum = 0.f;
#pragma unroll
        for (int k = 0; k < NK; ++k) { w[k] = __expf(w[k] - mxe); ssum += w[k]; }
        float inv = 1.f / ssum;
        for (int c = lane; c < NCH; c += 32) {
            float a = 0.f;
#pragma unroll
            for (int k = 0; k < NK; ++k)
                a += w[k] * hw[(long long)jg[k] * ND + h * NCH + c];
            float v = a * inv + bias[h * NCH + c];
            out[(long long)node * ND + h * NCH + c] = v > 0.f ? v : 0.f;
        }
    }
}

// row softmax (optionally with additive mask row = mask[(row % mask_mod)])
__global__ __launch_bounds__(256)
void softmax_kernel(float* __restrict__ S, int cols,
                    const float* __restrict__ mask, int mask_mod, long long mask_ld)
{
    __shared__ float red[256];
    long long row = blockIdx.x;
    float* p = S + row * cols;
    const float* mrow = mask ? mask + (long long)(row % mask_mod) * mask_ld : nullptr;

    float mx = -3.0e38f;
    for (int c = threadIdx.x; c < cols; c += 256) {
        float v = p[c] + (mrow ? mrow[c] : 0.f);
        mx = fmaxf(mx, v);
    }
    mx = blk_reduce_max(mx, red);

    float sum = 0.f;
    for (int c = threadIdx.x; c < cols; c += 256) {
        float v = __expf(p[c] + (mrow ? mrow[c] : 0.f) - mx);
        p[c] = v; sum += v;
    }
    sum = blk_reduce_sum(sum, red);
    float inv = 1.f / sum;
    for (int c = threadIdx.x; c < cols; c += 256) p[c] *= inv;
}

// out = LayerNorm(a (+ b), gamma, beta); out may alias a
__global__ __launch_bounds__(256)
void layernorm_kernel(const float* __restrict__ a, const float* __restrict__ b,
                      const float* __restrict__ g, const float* __restrict__ be,
                      float* __restrict__ out, int cols)
{
    __shared__ float red[256];
    long long row = blockIdx.x;
    const float* pa = a + row * cols;
    const float* pb = b ? b + row * cols : nullptr;

    float s = 0.f;
    for (int c = threadIdx.x; c < cols; c += 256)
        s += pa[c] + (pb ? pb[c] : 0.f);
    float mean = blk_reduce_sum(s, red) / cols;

    float vs = 0.f;
    for (int c = threadIdx.x; c < cols; c += 256) {
        float d = pa[c] + (pb ? pb[c] : 0.f) - mean;
        vs += d * d;
    }
    float var = blk_reduce_sum(vs, red) / cols;
    float inv = rsqrtf(var + 1e-5f);

    for (int c = threadIdx.x; c < cols; c += 256) {
        float v = pa[c] + (pb ? pb[c] : 0.f);
        out[row * cols + c] = (v - mean) * inv * g[c] + be[c];
    }
}

// gfeat[b,d] = mean over N of h[b,:,d]
__global__ __launch_bounds__(256)
void meanpool_kernel(const float* __restrict__ h, float* __restrict__ g)
{
    int b = blockIdx.x, d = threadIdx.x;
    float s = 0.f;
    for (int n = 0; n < NN; ++n) s += h[((long long)b * NN + n) * ND + d];
    g[b * ND + d] = s * (1.f / NN);
}

// z[b] = concat(gfeat[b], pooled[b] flattened)
__global__ __launch_bounds__(256)
void concat_kernel(const float* __restrict__ gf, const float* __restrict__ pooled,
                   float* __restrict__ z)
{
    int i = blockIdx.x * 256 + threadIdx.x;      // over NB * (ND + NNQ*ND)
    const int CZ = ND + NNQ * ND;                // 1280
    if (i >= NB * CZ) return;
    int b = i / CZ, c = i % CZ;
    z[i] = (c < ND) ? gf[b * ND + c] : pooled[b * NNQ * ND + (c - ND)];
}

// ---------------------------------------------------------------------------
// host-side launch helpers
// ---------------------------------------------------------------------------
static void gemm_full(hipStream_t st,
                      const float* A, long long lda, long long sAo, long long sAi,
                      const float* Bp, long long ldb, long long sBo, long long sBi,
                      float* C, long long ldc, long long sCo, long long sCi,
                      const float* bias, const float* addend, long long ldadd,
                      int M, int Nn, int K, int nOuter, int nInner,
                      float alpha, bool transB, bool relu)
{
    dim3 grid((Nn + TN - 1) / TN, (M + TM - 1) / TM, nOuter * nInner);
    wmma_gemm_kernel<<<grid, dim3(256), 0, st>>>(
        A, lda, sAo, sAi, Bp, ldb, sBo, sBi, C, ldc, sCo, sCi,
        bias, addend, ldadd, M, Nn, K, nInner, alpha, transB ? 1 : 0, relu ? 1 : 0);
}

// C[M,Nn] = A[M,K] @ W[Nn,K]^T (+bias)(+addend)(relu), C row stride ldc
static void gemm_lin(hipStream_t st, const float* A, int K, const float* W,
                     float* C, int M, int Nn, const float* bias,
                     bool relu = false, const float* addend = nullptr,
                     int ldadd = 0, int ldc = -1)
{
    if (ldc < 0) ldc = Nn;
    gemm_full(st, A, K, 0, 0, W, K, 0, 0, C, ldc, 0, 0,
              bias, addend, ldadd, M, Nn, K, 1, 1, 1.f, true, relu);
}

// ---------------------------------------------------------------------------
extern "C" void kernel_launch(void* const* d_in, const int* in_sizes, int n_in,
                              void* d_out, int out_size, void* d_ws, size_t ws_size,
                              hipStream_t stream)
{
    auto in = [&](int i) { return (const float*)d_in[i]; };

    // --- input indices (setup_inputs dict order, depth-first) ---
    const float* feats     = in(0);   // [8,1024,512]
    const float* cellposes = in(1);   // [8,1024,2]
    // in(2) = masks (unused by reference math)
    const float* attn_mask = in(3);   // [4,1024]
    const float* embW = in(4);  const float* embb = in(5);
    const float* xt = in(6);  const float* x1W = in(7);  const float* x1b = in(8);
    const float* x2W = in(9); const float* x2b = in(10);
    const float* yt = in(11); const float* y1W = in(12); const float* y1b = in(13);
    const float* y2W = in(14); const float* y2b = in(15);
    // gat[i]: base 16 + 4*i -> W,b,as,ad
    // trans[i]: base 24 + 12*i -> inW,inb,outW,outb,l1W,l1b,l2W,l2b,ln1g,ln1b,ln2g,ln2b
    const float* pinW  = in(48); const float* pinb  = in(49);
    const float* poutW = in(50); const float* poutb = in(51);
    const float* qe  = in(52);
    const float* lng = in(53); const float* lnb = in(54);
    const float* W1 = in(55); const float* b1 = in(56);
    const float* g1 = in(57); const float* b1n = in(58);
    const float* W2 = in(59); const float* b2 = in(60);
    const float* g2 = in(61); const float* b2n = in(62);
    const float* W3 = in(63); const float* b3 = in(64);
    const float* c0W = in(65); const float* c0b = in(66);
    const float* c1W = in(67); const float* c1b = in(68);
    const float* c2W = in(69); const float* c2b = in(70);
    const float* c3W = in(71); const float* c3b = in(72);

    float* out = (float*)d_out;   // final [8,1024] then logits [8,24]

    // --- workspace carving (floats) ---
    float* w = (float*)d_ws;
    size_t off = 0;
    auto alloc = [&](size_t n) { float* p = w + off; off += n; return p; };
    float* posemb = alloc((size_t)NODES * ND);        // 2M
    float* h      = alloc((size_t)NODES * ND);        // 2M
    float* tmp    = alloc((size_t)NODES * ND);        // 2M
    float* hwb    = alloc((size_t)NODES * ND);        // 2M
    float* qkv    = alloc((size_t)NODES * 3 * ND);    // 6M (aliased as kv later)
    float* Sbuf   = alloc((size_t)NB * NH * NN * NN); // 16M (aliased as ffn1/PE scratch)
    float* attnb  = alloc((size_t)NODES * ND);        // 2M
    float* qbuf   = alloc((size_t)NNQ * ND);
    float* Spool  = alloc((size_t)NB * NH * NNQ * NN);
    float* poolpre= alloc((size_t)NB * NNQ * ND);
    float* pooledf= alloc((size_t)NB * NNQ * ND);
    float* gfeatb = alloc((size_t)NB * ND);
    float* zb     = alloc((size_t)NB * (ND + NNQ * ND));
    float* z1     = alloc((size_t)NB * 1024);
    float* z2     = alloc((size_t)NB * 1024);
    float* cb0    = alloc((size_t)NB * 1032);
    float* cb1    = alloc((size_t)NB * 512);
    float* cb2    = alloc((size_t)NB * 256);
    float* esb    = alloc((size_t)NODES * NH);
    float* edb    = alloc((size_t)NODES * NH);
    float* statsb = alloc(32);
    int*   nbrp   = (int*)alloc((size_t)NODES * NK);
    (void)off; (void)ws_size; (void)in_sizes; (void)n_in; (void)out_size;

    // PE scratch aliases the (not-yet-used) attention score buffer
    float* xg    = Sbuf;
    float* yg    = Sbuf + (size_t)NODES * NCH;
    float* petmp = Sbuf + (size_t)2 * NODES * NCH;
    float* ffn1  = Sbuf;      // FFN hidden aliases score buffer (disjoint in time)
    float* kv    = qkv;       // pooling K/V aliases qkv (layers done by then)

    const float iscale = 0.08838834764831845f;  // 1/sqrt(128)

    // ---- spatial kNN graph ----
    pos_stats_kernel<<<NB, 256, 0, stream>>>(cellposes, statsb);
    knn_kernel<<<dim3(NN / 256, NB), 256, 0, stream>>>(cellposes, statsb, nbrp);

    // ---- 2D position embedding ----
    pe_gather_kernel<<<NODES, 128, 0, stream>>>(cellposes, xt, yt, xg, yg);
    gemm_lin(stream, xg, NCH, x1W, petmp, NODES, NCH, x1b, true);
    gemm_lin(stream, petmp, NCH, x2W, posemb + 0,   NODES, NCH, x2b, false, nullptr, 0, ND);
    gemm_lin(stream, yg, NCH, y1W, petmp, NODES, NCH, y1b, true);
    gemm_lin(stream, petmp, NCH, y2W, posemb + NCH, NODES, NCH, y2b, false, nullptr, 0, ND);

    // ---- input embedding: h = feats @ embW^T + embb + posemb ----
    gemm_lin(stream, feats, NE, embW, h, NODES, ND, embb, false, posemb, ND);

    // ---- GAT layers ----
    for (int i = 0; i < 2; ++i) {
        const float* gW  = in(16 + 4 * i + 0);
        const float* gb  = in(16 + 4 * i + 1);
        const float* gas = in(16 + 4 * i + 2);
        const float* gad = in(16 + 4 * i + 3);
        gemm_lin(stream, h, ND, gW, hwb, NODES, ND, nullptr);
        gat_score_kernel<<<(NODES * NH) / 256, 256, 0, stream>>>(hwb, gas, gad, esb, edb);
        gat_aggregate_kernel<<<NODES / 8, 256, 0, stream>>>(hwb, esb, edb, nbrp, gb, h);
    }

    // ---- transformer encoder layers (post-norm, relu FFN) ----
    for (int i = 0; i < 2; ++i) {
        const int tb = 24 + 12 * i;
        const float* inW  = in(tb + 0); const float* inb  = in(tb + 1);
        const float* outW = in(tb + 2); const float* outb = in(tb + 3);
        const float* l1W  = in(tb + 4); const float* l1b  = in(tb + 5);
        const float* l2W  = in(tb + 6); const float* l2b  = in(tb + 7);
        const float* ln1g = in(tb + 8); const float* ln1b = in(tb + 9);
        const float* ln2g = in(tb + 10); const float* ln2b = in(tb + 11);

        // QKV in-projection: [NODES, 768]
        gemm_lin(stream, h, ND, inW, qkv, NODES, 3 * ND, inb);

        // scores S[b,h] = Q @ K^T / sqrt(hd)   (batched over 16 = 8*2)
        gemm_full(stream,
                  qkv,            3 * ND, (long long)NN * 3 * ND, NHD,   // Q
                  qkv + ND,       3 * ND, (long long)NN * 3 * ND, NHD,   // K
                  Sbuf,           NN, (long long)2 * NN * NN, (long long)NN * NN,
                  nullptr, nullptr, 0,
                  NN, NN, NHD, NB, NH, iscale, true, false);
        softmax_kernel<<<NB * NH * NN, 256, 0, stream>>>(Sbuf, NN, nullptr, 1, 0);
        // O[b,h] = S @ V  -> interleave heads into attnb [NODES, 256]
        gemm_full(stream,
                  Sbuf,           NN, (long long)2 * NN * NN, (long long)NN * NN,
                  qkv + 2 * ND,   3 * ND, (long long)NN * 3 * ND, NHD,   // V ([K,N])
                  attnb,          ND, (long long)NN * ND, NHD,
                  nullptr, nullptr, 0,
                  NN, NHD, NN, NB, NH, 1.f, false, false);
        // out projection + residual LN
        gemm_lin(stream, attnb, ND, outW, tmp, NODES, ND, outb);
        layernorm_kernel<<<NODES, 256, 0, stream>>>(h, tmp, ln1g, ln1b, h, ND);
        // FFN
        gemm_lin(stream, h, ND, l1W, ffn1, NODES, NHID, l1b, true);
        gemm_lin(stream, ffn1, NHID, l2W, tmp, NODES, ND, l2b);
        layernorm_kernel<<<NODES, 256, 0, stream>>>(h, tmp, ln2g, ln2b, h, ND);
    }

    // ---- sparse-attention query pooling ----
    gemm_lin(stream, qe, ND, pinW, qbuf, NNQ, ND, pinb);                 // q in-proj
    gemm_lin(stream, h, ND, pinW + ND * ND, kv, NODES, 2 * ND, pinb + ND); // k|v
    gemm_full(stream,
              qbuf,      ND, 0, NHD,                                   // shared Q
              kv,        2 * ND, (long long)NN * 2 * ND, NHD,          // K
              Spool,     NN, (long long)NH * NNQ * NN, (long long)NNQ * NN,
              nullptr, nullptr, 0,
              NNQ, NN, NHD, NB, NH, iscale, true, false);
    softmax_kernel<<<NB * NH * NNQ, 256, 0, stream>>>(Spool, NN, attn_mask, NNQ, NN);
    gemm_full(stream,
              Spool,     NN, (long long)NH * NNQ * NN, (long long)NNQ * NN,
              kv + ND,   2 * ND, (long long)NN * 2 * ND, NHD,          // V
              poolpre,   ND, (long long)NNQ * ND, NHD,
              nullptr, nullptr, 0,
              NNQ, NHD, NN, NB, NH, 1.f, false, false);
    gemm_lin(stream, poolpre, ND, poutW, pooledf, NB * NNQ, ND, poutb);
    layernorm_kernel<<<NB * NNQ, 256, 0, stream>>>(pooledf, nullptr, lng, lnb, pooledf, ND);

    // ---- head MLP ----
    meanpool_kernel<<<NB, ND, 0, stream>>>(h, gfeatb);
    const int CZ = ND + NNQ * ND;  // 1280
    concat_kernel<<<(NB * CZ + 255) / 256, 256, 0, stream>>>(gfeatb, pooledf, zb);
    gemm_lin(stream, zb, CZ, W1, z1, NB, 1024, b1, true);
    layernorm_kernel<<<NB, 256, 0, stream>>>(z1, nullptr, g1, b1n, z1, 1024);
    gemm_lin(stream, z1, 1024, W2, z2, NB, 1024, b2, true);
    layernorm_kernel<<<NB, 256, 0, stream>>>(z2, nullptr, g2, b2n, z2, 1024);
    gemm_lin(stream, z2, 1024, W3, out, NB, NOUT, b3);                 // final -> d_out

    // ---- classifier ----
    gemm_lin(stream, out, NOUT, c0W, cb0, NB, 1032, c0b, true);
    gemm_lin(stream, cb0, 1032, c1W, cb1, NB, 512, c1b, true);
    gemm_lin(stream, cb1, 512,  c2W, cb2, NB, 256, c2b, true);
    gemm_lin(stream, cb2, 256,  c3W, out + (size_t)NB * NOUT, NB, NCLS, c3b); // logits
}